// BaseSegmentTree_17420387352878
// MI455X (gfx1250) — compile-verified
//
#include <hip/hip_runtime.h>
#include <math.h>

typedef __attribute__((ext_vector_type(2))) float v2f;
typedef __attribute__((ext_vector_type(8))) float v8f;

#define DIMD   128
#define LEAFN  1024
#define NNODE  2048
#define NLAYER 2

static inline unsigned cdiv(unsigned a, unsigned b) { return (a + b - 1) / b; }

// ---------------- tree build ----------------
__global__ void copy_leaves_kernel(const float* __restrict__ elements,
                                   float* __restrict__ nf, int B) {
  int idx = blockIdx.x * blockDim.x + threadIdx.x;        // over B*LEAFN*DIMD
  int total = B * LEAFN * DIMD;
  if (idx >= total) return;
  int d = idx % DIMD;
  int j = (idx / DIMD) % LEAFN;
  int b = idx / (DIMD * LEAFN);
  nf[(size_t)b * NNODE * DIMD + (size_t)(LEAFN + j) * DIMD + d] = elements[idx];
}

__global__ void reduce_level_kernel(float* __restrict__ nf, int B, int s, int c) {
  int idx = blockIdx.x * blockDim.x + threadIdx.x;        // over B*c*DIMD
  int total = B * c * DIMD;
  if (idx >= total) return;
  int d = idx % DIMD;
  int k = (idx / DIMD) % c;
  int b = idx / (DIMD * c);
  int i = s + k;
  size_t base = (size_t)b * NNODE * DIMD;
  nf[base + (size_t)i * DIMD + d] =
      0.5f * (nf[base + (size_t)(2 * i) * DIMD + d] +
              nf[base + (size_t)(2 * i + 1) * DIMD + d]);
}

// node_feat + positional encoding -> x  (x layout identical to nf: [B, N, D])
__global__ void x_init_kernel(const float* __restrict__ nf,
                              float* __restrict__ x, int B) {
  int idx = blockIdx.x * blockDim.x + threadIdx.x;        // over B*NNODE*DIMD
  int total = B * NNODE * DIMD;
  if (idx >= total) return;
  int d = idx % DIMD;
  int i = (idx / DIMD) % NNODE;
  float val = (i == 0) ? -1.0f : nf[idx];
  // positional encoding: vpos = floor(log2(i)) (i==0 -> -1), hpos = i - 2^vpos
  float vpos, hpos;
  if (i == 0) { vpos = -1.0f; hpos = -0.5f; }
  else {
    int v = 31 - __clz(i);                 // exact floor(log2) via integers
    vpos = (float)v;
    hpos = (float)(i - (1 << v));
  }
  float pos = (d < 64) ? hpos : vpos;
  int dd = (d < 64) ? d : d - 64;
  int j = dd >> 1;
  // inv[j] = exp(-j * ln(10000)/32)
  float inv = expf(-(float)j * 0.28782313662425574f);
  float ang = pos * inv;
  float e = (dd & 1) ? cosf(ang) : sinf(ang);
  x[idx] = val + e;
}

// ---------------- CSR row bounds (dst is globally non-decreasing) ----------------
__global__ void rows_clear_kernel(int* __restrict__ rowStart,
                                  int* __restrict__ rowEnd, int R) {
  int n = blockIdx.x * blockDim.x + threadIdx.x;
  if (n >= R) return;
  rowStart[n] = 0;
  rowEnd[n] = 0;
}

__global__ void rows_bound_kernel(const int* __restrict__ dst, int E,
                                  int* __restrict__ rowStart,
                                  int* __restrict__ rowEnd) {
  int i = blockIdx.x * blockDim.x + threadIdx.x;
  if (i >= E) return;
  int dv = dst[i];
  if (i == 0 || dst[i - 1] != dv) rowStart[dv] = i;
  if (i == E - 1 || dst[i + 1] != dv) rowEnd[dv] = i + 1;
}

// ---------------- LayerNorm + exact GELU (one wave32 per row) ----------------
__global__ __launch_bounds__(256) void ln_gelu_kernel(
    const float* __restrict__ x, const float* __restrict__ gamma,
    const float* __restrict__ beta, float* __restrict__ h, int R) {
  int row = blockIdx.x * (blockDim.x >> 5) + (threadIdx.x >> 5);
  int lane = threadIdx.x & 31;
  if (row >= R) return;                    // uniform per wave
  const float4* xr = (const float4*)(x + (size_t)row * DIMD);
  float4 v = xr[lane];                     // elems lane*4 .. lane*4+3
  float s = v.x + v.y + v.z + v.w;
  #pragma unroll
  for (int off = 16; off >= 1; off >>= 1) s += __shfl_xor(s, off, 32);
  float mu = s * (1.0f / 128.0f);
  float d0 = v.x - mu, d1 = v.y - mu, d2 = v.z - mu, d3 = v.w - mu;
  float q = d0 * d0 + d1 * d1 + d2 * d2 + d3 * d3;
  #pragma unroll
  for (int off = 16; off >= 1; off >>= 1) q += __shfl_xor(q, off, 32);
  float rs = rsqrtf(q * (1.0f / 128.0f) + 1e-5f);
  int base = lane * 4;
  float g0 = gamma[base + 0], g1 = gamma[base + 1], g2 = gamma[base + 2], g3 = gamma[base + 3];
  float b0 = beta[base + 0], b1 = beta[base + 1], b2 = beta[base + 2], b3 = beta[base + 3];
  float t0 = d0 * rs * g0 + b0;
  float t1 = d1 * rs * g1 + b1;
  float t2 = d2 * rs * g2 + b2;
  float t3 = d3 * rs * g3 + b3;
  const float k = 0.70710678118654752f;    // 1/sqrt(2)
  float4 o;
  o.x = 0.5f * t0 * (1.0f + erff(t0 * k));
  o.y = 0.5f * t1 * (1.0f + erff(t1 * k));
  o.z = 0.5f * t2 * (1.0f + erff(t2 * k));
  o.w = 0.5f * t3 * (1.0f + erff(t3 * k));
  ((float4*)(h + (size_t)row * DIMD))[lane] = o;
}

// ---------------- deterministic mean aggregation (one block per dst node) ----------------
__global__ __launch_bounds__(128) void aggregate_kernel(
    const float* __restrict__ h, const int* __restrict__ src,
    const int* __restrict__ rowStart, const int* __restrict__ rowEnd,
    float* __restrict__ agg) {
  int n = blockIdx.x;
  int t = threadIdx.x;                     // 0..127, one feature dim per thread
  int rs = rowStart[n], re = rowEnd[n];
  __shared__ int sIdx[128];
  float acc = 0.0f;
  for (int b0 = rs; b0 < re; b0 += 128) {
    int m = re - b0; if (m > 128) m = 128;
    __syncthreads();
    if (t < m) sIdx[t] = src[b0 + t];
    __syncthreads();
    for (int e = 0; e < m; ++e) acc += h[(size_t)sIdx[e] * DIMD + t];
  }
  int cnt = re - rs;
  agg[(size_t)n * DIMD + t] = (cnt > 0) ? acc / (float)cnt : 0.0f;
}

// ---------------- fused dual GEMM + bias + residual via fp32 WMMA ----------------
// x[m, n] += sum_k agg[m,k]*Wn[k,n] + sum_k h[m,k]*Wr[k,n] + bn[n]
// One wave per 16x16 tile; 8 waves per block cover the full N=128.
__global__ __launch_bounds__(256) void gemm_resid_kernel(
    float* __restrict__ x, const float* __restrict__ agg,
    const float* __restrict__ h, const float* __restrict__ wn,
    const float* __restrict__ bn, const float* __restrict__ wr) {
  int lane = threadIdx.x & 31;
  int wv = threadIdx.x >> 5;               // 0..7 -> N tile
  int m0 = blockIdx.x * 16;
  int n0 = wv * 16;
  int mrow = lane & 15;                    // A-fragment row (ISA: lanes 0-15 & 16-31 both M=0..15)
  int kk = (lane >> 4) << 1;               // K offset: 0 (lanes 0-15) or 2 (lanes 16-31)
  int ncol = lane & 15;                    // B/C/D column within tile

  v8f acc = {0.f, 0.f, 0.f, 0.f, 0.f, 0.f, 0.f, 0.f};
  const float* Arow0 = agg + (size_t)(m0 + mrow) * DIMD;
  const float* Arow1 = h + (size_t)(m0 + mrow) * DIMD;

  #pragma unroll 4
  for (int k0 = 0; k0 < DIMD; k0 += 4) {
    v2f a = *(const v2f*)(Arow0 + k0 + kk);
    v2f b;
    b.x = wn[(size_t)(k0 + kk) * DIMD + n0 + ncol];
    b.y = wn[(size_t)(k0 + kk + 1) * DIMD + n0 + ncol];
    acc = __builtin_amdgcn_wmma_f32_16x16x4_f32(false, a, false, b, (short)0,
                                                acc, false, false);
  }
  #pragma unroll 4
  for (int k0 = 0; k0 < DIMD; k0 += 4) {
    v2f a = *(const v2f*)(Arow1 + k0 + kk);
    v2f b;
    b.x = wr[(size_t)(k0 + kk) * DIMD + n0 + ncol];
    b.y = wr[(size_t)(k0 + kk + 1) * DIMD + n0 + ncol];
    acc = __builtin_amdgcn_wmma_f32_16x16x4_f32(false, a, false, b, (short)0,
                                                acc, false, false);
  }

  float bias = bn[n0 + ncol];
  int rbase = m0 + ((lane >> 4) << 3);     // C/D: VGPR r -> M = r (+8 for lanes 16-31)
  #pragma unroll
  for (int r = 0; r < 8; ++r) {
    size_t off = (size_t)(rbase + r) * DIMD + n0 + ncol;
    x[off] += acc[r] + bias;
  }
}

extern "C" void kernel_launch(void* const* d_in, const int* in_sizes, int n_in,
                              void* d_out, int out_size, void* d_ws, size_t ws_size,
                              hipStream_t stream) {
  const float* elements = (const float*)d_in[0];  // [B, LEAF, D]
  const float* ln_gamma = (const float*)d_in[1];  // [L, D]
  const float* ln_beta  = (const float*)d_in[2];  // [L, D]
  const float* w_nei    = (const float*)d_in[3];  // [L, D, D]
  const float* b_nei    = (const float*)d_in[4];  // [L, D]
  const float* w_root   = (const float*)d_in[5];  // [L, D, D]
  const int*   eidx     = (const int*)d_in[6];    // [2, E] flat

  int B = in_sizes[0] / (LEAFN * DIMD);           // 8
  int R = B * NNODE;                              // 16384 rows
  int E = in_sizes[6] / 2;
  const int* srcArr = eidx;
  const int* dstArr = eidx + E;

  float* x = (float*)d_out;                       // [R, D], used in place across layers

  // workspace layout
  float* nf  = (float*)d_ws;                      // [B, N, D]
  float* h   = nf + (size_t)R * DIMD;             // [R, D]
  float* agg = h + (size_t)R * DIMD;              // [R, D]
  int* rowStart = (int*)(agg + (size_t)R * DIMD); // [R]
  int* rowEnd   = rowStart + R;                   // [R]

  // ---- build tree node features ----
  {
    unsigned tot = (unsigned)(B * LEAFN * DIMD);
    copy_leaves_kernel<<<cdiv(tot, 256), 256, 0, stream>>>(elements, nf, B);
  }
  for (int v = 9; v >= 0; --v) {
    int s = 1 << v, c = 1 << v;
    unsigned tot = (unsigned)(B * c * DIMD);
    reduce_level_kernel<<<cdiv(tot, 256), 256, 0, stream>>>(nf, B, s, c);
  }
  {
    unsigned tot = (unsigned)(B * NNODE * DIMD);
    x_init_kernel<<<cdiv(tot, 256), 256, 0, stream>>>(nf, x, B);
  }

  // ---- CSR row bounds from sorted dst ----
  rows_clear_kernel<<<cdiv((unsigned)R, 256), 256, 0, stream>>>(rowStart, rowEnd, R);
  rows_bound_kernel<<<cdiv((unsigned)E, 256), 256, 0, stream>>>(dstArr, E, rowStart, rowEnd);

  // ---- L GCN layers ----
  for (int l = 0; l < NLAYER; ++l) {
    ln_gelu_kernel<<<cdiv((unsigned)R, 8), 256, 0, stream>>>(
        x, ln_gamma + (size_t)l * DIMD, ln_beta + (size_t)l * DIMD, h, R);
    aggregate_kernel<<<(unsigned)R, 128, 0, stream>>>(h, srcArr, rowStart, rowEnd, agg);
    gemm_resid_kernel<<<(unsigned)(R / 16), 256, 0, stream>>>(
        x, agg, h, w_nei + (size_t)l * DIMD * DIMD, b_nei + (size_t)l * DIMD,
        w_root + (size_t)l * DIMD * DIMD);
  }
}